// CumulativeGroupNormModel_46823733461777
// MI455X (gfx1250) — compile-verified
//
#include <hip/hip_runtime.h>

typedef __attribute__((ext_vector_type(2))) float v2f;
typedef __attribute__((ext_vector_type(8))) float v8f;

#define CCH 512
#define TFR 8192
#define EPSV 1e-5f

// ---------------------------------------------------------------------------
// Kernel 1: per-frame s1 = sum_c x, s2 = sum_c x^2 via V_WMMA_F32_16X16X4_F32
// with B = all-ones (every column of D becomes the row-sum of A).
// One wave handles 16 consecutive frames (M), reducing C in chunks of K=4.
// A layout (16x4 f32): lane L holds A[M = L&15, K = (L>>4)*2 + {0,1}].
// D layout (16x16 f32): vgpr r: lanes 0-15 = row r, lanes 16-31 = row r+8.
// ---------------------------------------------------------------------------
__global__ __launch_bounds__(128) void cgn_stats_kernel(const float* __restrict__ x,
                                                        float* __restrict__ s1out,
                                                        float* __restrict__ s2out) {
    const int lane = threadIdx.x & 31;
    const int waveInBlock = threadIdx.x >> 5;
    const int tile = blockIdx.x * 4 + waveInBlock;   // 16 frames per tile
    const size_t f0 = (size_t)tile * 16;

    const int m  = lane & 15;
    const int kb = (lane >> 4) << 1;                 // 0 or 2

    const float* row = x + (f0 + (size_t)m) * CCH + kb;

    v2f ones; ones.x = 1.0f; ones.y = 1.0f;
    v8f acc1 = {};
    v8f acc2 = {};

    #pragma unroll 4
    for (int c0 = 0; c0 < CCH; c0 += 4) {
        v2f a = *(const v2f*)(row + c0);
        v2f sq; sq.x = a.x * a.x; sq.y = a.y * a.y;
        // D = A x Ones + C  -> every column of D is the row-sum of A
        acc1 = __builtin_amdgcn_wmma_f32_16x16x4_f32(
            false, a,  false, ones, (short)0, acc1, false, false);
        acc2 = __builtin_amdgcn_wmma_f32_16x16x4_f32(
            false, sq, false, ones, (short)0, acc2, false, false);
    }

    // Lane 0 holds rows 0..7 in vgprs 0..7; lane 16 holds rows 8..15.
    if (lane == 0) {
        #pragma unroll
        for (int r = 0; r < 8; ++r) {
            s1out[f0 + r] = acc1[r];
            s2out[f0 + r] = acc2[r];
        }
    }
    if (lane == 16) {
        #pragma unroll
        for (int r = 0; r < 8; ++r) {
            s1out[f0 + 8 + r] = acc1[r];
            s2out[f0 + 8 + r] = acc2[r];
        }
    }
}

// ---------------------------------------------------------------------------
// Kernel 2: one block (1024 threads) per batch. Two block-wide inclusive
// scans over T=8192: cumsum(s1) -> mean[t], then cumsum(d) -> rstd[t].
// In-place: s1 buffer becomes mean, s2 buffer becomes rstd.
// ---------------------------------------------------------------------------
__global__ __launch_bounds__(1024) void cgn_scan_kernel(float* __restrict__ s1_mean,
                                                        float* __restrict__ s2_rstd) {
    __shared__ float sh[1024];
    const int tid = threadIdx.x;
    const int PER = TFR / 1024;                      // 8 frames per thread
    const size_t base = (size_t)blockIdx.x * TFR + (size_t)tid * PER;

    float rs1[8], rs2[8], p1[8];
    float run = 0.f;
    #pragma unroll
    for (int i = 0; i < PER; ++i) {
        rs1[i] = s1_mean[base + i];
        rs2[i] = s2_rstd[base + i];
        run += rs1[i];
        p1[i] = run;                                 // local inclusive prefix
    }
    sh[tid] = run;
    __syncthreads();
    for (int off = 1; off < 1024; off <<= 1) {
        float v = sh[tid];
        float a = (tid >= off) ? sh[tid - off] : 0.f;
        __syncthreads();
        sh[tid] = v + a;
        __syncthreads();
    }
    const float excl = sh[tid] - run;                // exclusive prefix of thread totals

    float p2[8], meanv[8];
    float run2 = 0.f;
    #pragma unroll
    for (int i = 0; i < PER; ++i) {
        float cs1  = excl + p1[i];
        float cnt  = (float)(tid * PER + i + 1) * (float)CCH;
        float mean = cs1 / cnt;
        meanv[i] = mean;
        // sum_c (x - mean)^2 = s2 - 2*mean*s1 + C*mean^2
        float d = rs2[i] - 2.0f * mean * rs1[i] + (float)CCH * mean * mean;
        run2 += d;
        p2[i] = run2;
    }
    __syncthreads();
    sh[tid] = run2;
    __syncthreads();
    for (int off = 1; off < 1024; off <<= 1) {
        float v = sh[tid];
        float a = (tid >= off) ? sh[tid - off] : 0.f;
        __syncthreads();
        sh[tid] = v + a;
        __syncthreads();
    }
    const float excl2 = sh[tid] - run2;

    #pragma unroll
    for (int i = 0; i < PER; ++i) {
        float cnt = (float)(tid * PER + i + 1) * (float)CCH;
        float var = (excl2 + p2[i]) / cnt;
        s1_mean[base + i] = meanv[i];
        s2_rstd[base + i] = rsqrtf(var + EPSV);
    }
}

// ---------------------------------------------------------------------------
// Kernel 3: streaming normalize, float4-vectorized.
// out = (x - mean[f]) * rstd[f] * weight[c]
// ---------------------------------------------------------------------------
__global__ __launch_bounds__(256) void cgn_norm_kernel(const float4* __restrict__ x4,
                                                       const float4* __restrict__ w4,
                                                       const float* __restrict__ meanArr,
                                                       const float* __restrict__ rstdArr,
                                                       float4* __restrict__ out4,
                                                       long n4) {
    long idx = (long)blockIdx.x * blockDim.x + threadIdx.x;
    if (idx >= n4) return;
    long f = idx >> 7;                               // / (512/4)
    int  c = (int)(idx & 127);
    float mean = meanArr[f];
    float rstd = rstdArr[f];
    float4 xv = x4[idx];
    float4 wv = w4[c];
    float4 o;
    o.x = (xv.x - mean) * rstd * wv.x;
    o.y = (xv.y - mean) * rstd * wv.y;
    o.z = (xv.z - mean) * rstd * wv.z;
    o.w = (xv.w - mean) * rstd * wv.w;
    out4[idx] = o;
}

extern "C" void kernel_launch(void* const* d_in, const int* in_sizes, int n_in,
                              void* d_out, int out_size, void* d_ws, size_t ws_size,
                              hipStream_t stream) {
    const float* x = (const float*)d_in[0];
    const float* w = (const float*)d_in[1];
    float* out = (float*)d_out;

    const long total  = (long)in_sizes[0];           // B*T*C = 33,554,432
    const long frames = total / CCH;                 // B*T   = 65,536
    const int  B      = (int)(frames / TFR);         // 8

    float* ws0 = (float*)d_ws;                       // s1 -> mean  [frames]
    float* ws1 = ws0 + frames;                       // s2 -> rstd  [frames]

    // 1) per-frame sums via WMMA (16 frames per wave, 4 waves per block)
    const int tiles = (int)(frames / 16);            // 4096
    hipLaunchKernelGGL(cgn_stats_kernel, dim3(tiles / 4), dim3(128), 0, stream,
                       x, ws0, ws1);

    // 2) per-batch scans over T
    hipLaunchKernelGGL(cgn_scan_kernel, dim3(B), dim3(1024), 0, stream, ws0, ws1);

    // 3) streaming normalize
    const long n4 = total / 4;
    hipLaunchKernelGGL(cgn_norm_kernel, dim3((unsigned)((n4 + 255) / 256)), dim3(256), 0, stream,
                       (const float4*)x, (const float4*)w, ws0, ws1, (float4*)out, n4);
}